// PolynomialKernel_91147795955941
// MI455X (gfx1250) — compile-verified
//
#include <hip/hip_runtime.h>

// Polynomial feature kernel for MI455X (gfx1250, wave32).
// B=4, L=2048, D=128.  out[tok] = [ x_norm (128) | triu(x_norm ⊗ x_norm)*s (8256) ]
// Store-bandwidth bound (~275 MB writes -> ~12us floor @ 23.3 TB/s).
// Outer products computed with v_wmma_f32_16x16x4_f32 (effective K=1:
// A col K=0 / B row K=0 carry the 16-element vector slices, rest zero).

typedef __attribute__((ext_vector_type(2))) float v2f;
typedef __attribute__((ext_vector_type(4))) float v4f;
typedef __attribute__((ext_vector_type(8))) float v8f;

#define DIM 128
#define OUTD 8384          // 128 + 128*129/2
#define NTOK 8192          // B*L
#define WAVES_PER_BLOCK 8

// One 16x16 outer-product tile: D = a_col * b_row  (f32, full precision)
static __device__ __forceinline__ v8f outer_tile(float av, float bv) {
    v2f a = {av, 0.0f};    // A 16x4: K=0 col in lanes 0-15 of VGPR0; K=1..3 zero
    v2f b = {bv, 0.0f};    // B 4x16: K=0 row in lanes 0-15 of VGPR0; rest irrelevant
    v8f c = {};
    return __builtin_amdgcn_wmma_f32_16x16x4_f32(
        /*neg_a=*/false, a, /*neg_b=*/false, b,
        /*c_mod=*/(short)0, c, /*reuse_a=*/false, /*reuse_b=*/false);
}

__global__ __launch_bounds__(256) void poly_feat_wmma_kernel(
    const float* __restrict__ x,
    const float* __restrict__ degree_scales,
    float* __restrict__ out)
{
    __shared__ float lds_xn[WAVES_PER_BLOCK][DIM];

    const int tid  = threadIdx.x;
    const int wave = tid >> 5;
    const int lane = tid & 31;
    const int tok  = blockIdx.x * WAVES_PER_BLOCK + wave;   // grid sized exactly

    const float s2 = degree_scales[1];                      // uniform scalar

    // ---- load this token's 128 floats: 4 per lane (global_load_b128) ----
    const float* xt = x + (size_t)tok * DIM;
    v4f xv = *(const v4f*)(xt + lane * 4);

    // ---- wave32 sum-of-squares reduction ----
    float ss = xv.x * xv.x + xv.y * xv.y + xv.z * xv.z + xv.w * xv.w;
    #pragma unroll
    for (int m = 16; m >= 1; m >>= 1)
        ss += __shfl_xor(ss, m, 32);
    const float inv = 1.0f / fmaxf(sqrtf(ss), 1e-12f);

    v4f xn = xv * inv;

    // stage normalized vector in LDS (per-wave slice)
    float* xs = lds_xn[wave];
    *(v4f*)(xs + lane * 4) = xn;
    __syncthreads();   // uniform path; makes LDS writes visible across lanes

    // ---- degree-1 features: contiguous 128-float store ----
    float* outTok = out + (size_t)tok * OUTD;
    *(v4f*)(outTok + lane * 4) = xn;

    // ---- hoist WMMA operands into registers (no LDS in the tile loop) ----
    const int l15    = lane & 15;
    const int rowAdd = (lane >> 4) << 3;   // 0 for lanes 0-15, 8 for lanes 16-31

    float va[8], vb[8];
    #pragma unroll
    for (int t = 0; t < 8; ++t) {
        float tv = xs[t * 16 + l15];       // unconditional ds_load_b32
        tv = (lane < 16) ? tv : 0.0f;      // v_cndmask, not a branch
        va[t] = tv;                        // A operand (rows)
        vb[t] = tv * s2;                   // B operand (cols), pre-scaled
    }

    // ---- degree-2 via 36 WMMA outer-product tiles (fully unrolled) ----
    float* d2 = outTok + DIM;              // packed triu region

    #pragma unroll
    for (int ti = 0; ti < 8; ++ti) {
        // row base offsets 128*i - i*(i+1)/2, reused across all tj of this ti
        int rowbase[8];
        #pragma unroll
        for (int r = 0; r < 8; ++r) {
            const int i = ti * 16 + r + rowAdd;
            rowbase[r] = i * 128 - ((i * (i + 1)) >> 1);
        }

        // diagonal tile (needs j >= i predicate)
        {
            v8f c = outer_tile(va[ti], vb[ti]);
            const int j = ti * 16 + l15;
            #pragma unroll
            for (int r = 0; r < 8; ++r) {
                const int i = ti * 16 + r + rowAdd;
                if (j >= i)
                    d2[rowbase[r] + j] = c[r];
            }
        }

        // off-diagonal tiles: every element valid -> unconditional stores
        #pragma unroll
        for (int tj = ti + 1; tj < 8; ++tj) {
            v8f c = outer_tile(va[ti], vb[tj]);
            const int j = tj * 16 + l15;
            #pragma unroll
            for (int r = 0; r < 8; ++r)
                d2[rowbase[r] + j] = c[r];
        }
    }
}

extern "C" void kernel_launch(void* const* d_in, const int* in_sizes, int n_in,
                              void* d_out, int out_size, void* d_ws, size_t ws_size,
                              hipStream_t stream) {
    (void)in_sizes; (void)n_in; (void)out_size; (void)d_ws; (void)ws_size;
    const float* x  = (const float*)d_in[0];           // (4, 2048, 128) f32
    const float* ds = (const float*)d_in[1];           // (2,) f32 degree scales
    float* out      = (float*)d_out;                   // (4, 2048, 8384) f32

    dim3 grid(NTOK / WAVES_PER_BLOCK);                 // 1024 blocks
    dim3 block(256);                                   // 8 waves of 32
    hipLaunchKernelGGL(poly_feat_wmma_kernel, grid, block, 0, stream, x, ds, out);
}